// KNRMModel_77884936946320
// MI455X (gfx1250) — compile-verified
//
#include <hip/hip_runtime.h>
#include <hip/hip_bf16.h>
#include <math.h>

// ---------------------------------------------------------------------------
// KNRM on MI455X (gfx1250).
// Stage 1: gather + L2-normalize embedding rows into workspace (fp32).
// Stage 2: per-(b,c,h) 20x300x20 GEMM via V_WMMA_F32_16X16X4_F32 (fp32 WMMA:
//          the sigma=0.001 RBF kernel needs full fp32 fusion values), fused
//          with the RBF pooling / log / sum reduction so the 3.2M-element
//          fusion tensor never touches HBM.
//          Padded rows (20..31) are handled by a loop-invariant pointer
//          select onto a shared zero row, so the K-loop is branch-free and
//          fragment loads merge into ds_load_b64.
// Stage 3: attention + score + log_softmax epilogue (160 outputs).
// ---------------------------------------------------------------------------

#define B_   32
#define CDD_ 5
#define HIS_ 50
#define L_   20
#define D_   300
#define K_   20
#define RS_  304   // global workspace row stride (floats, 16B-aligned rows)
#define LS_  308   // LDS row stride (floats): 16B-aligned rows, conflict-free

typedef __attribute__((ext_vector_type(2))) float v2f;
typedef __attribute__((ext_vector_type(8))) float v8f;

// ---------------- Stage 1: gather + normalize --------------------------------
__global__ void knrm_normalize(const int* __restrict__ cand,
                               const int* __restrict__ clk,
                               const float* __restrict__ emb,
                               float* __restrict__ cddN,
                               float* __restrict__ hisN)
{
    const int NC = B_ * CDD_ * L_;       // 3200
    const int NH = B_ * HIS_ * L_;       // 32000
    int wavesPerBlock = blockDim.x >> 5;
    int wave = blockIdx.x * wavesPerBlock + (threadIdx.x >> 5);
    int lane = threadIdx.x & 31;
    if (wave >= NC + NH) return;

    int token;
    float* dst;
    if (wave < NC) { token = cand[wave];      dst = cddN + (size_t)wave * RS_; }
    else           { int r = wave - NC;
                     token = clk[r];          dst = hisN + (size_t)r   * RS_; }

    const float* src = emb + (size_t)token * D_;
    __builtin_prefetch(src, 0, 3);           // global_prefetch_b8

    float ss = 0.0f;
    for (int j = lane; j < D_; j += 32) { float v = src[j]; ss += v * v; }
    #pragma unroll
    for (int m = 16; m >= 1; m >>= 1) ss += __shfl_xor(ss, m, 32);
    float inv = 1.0f / fmaxf(sqrtf(ss), 1e-12f);
    for (int j = lane; j < D_; j += 32) dst[j] = src[j] * inv;
}

// ---------------- Stage 2: WMMA fusion + RBF pooling -------------------------
// One 128-thread block (4 waves) per (b,c,h). Waves own the 2x2 grid of 16x16
// C tiles covering the zero-padded 32x32 fusion matrix; K=300 in steps of 4.
__global__ void __launch_bounds__(128)
knrm_fusion_pool(const float* __restrict__ cddN,
                 const float* __restrict__ hisN,
                 const float* __restrict__ cpad,
                 const float* __restrict__ hpad,
                 float* __restrict__ pv)
{
    __shared__ __align__(16) float sC[L_][LS_];
    __shared__ __align__(16) float sH[L_][LS_];
    __shared__ __align__(16) float zrow[LS_];     // shared zero row for padding
    __shared__ float fus[32][33];
    __shared__ float psum[L_ * K_];
    __shared__ float sHP[L_];
    __shared__ float sCP[L_];

    const int blk = blockIdx.x;          // ((b*CDD)+c)*HIS + h
    const int h   = blk % HIS_;
    const int bc  = blk / HIS_;          // b*CDD + c
    const int b   = bc / CDD_;
    const int t   = threadIdx.x;

    // Cooperative float4 load of the two 20x300 operand panels into LDS.
    const float* cRow = cddN + (size_t)bc * L_ * RS_;
    const float* hRow = hisN + (size_t)(b * HIS_ + h) * L_ * RS_;
    const int V = D_ / 4;                // 75 float4 per row
    for (int i = t; i < L_ * V; i += 128) {
        int r = i / V, j = i % V;
        float4 vc = ((const float4*)(cRow + (size_t)r * RS_))[j];
        float4 vh = ((const float4*)(hRow + (size_t)r * RS_))[j];
        ((float4*)&sC[r][0])[j] = vc;
        ((float4*)&sH[r][0])[j] = vh;
    }
    for (int i = t; i < LS_; i += 128) zrow[i] = 0.0f;
    if (t < L_) {
        sCP[t] = cpad[bc * L_ + t];
        sHP[t] = hpad[(b * HIS_ + h) * L_ + t];
    }
    __syncthreads();

    // --- fp32 WMMA: D(16x16) += A(16x4) * B(4x16), K marched by 4 ---
    const int lane = t & 31;
    const int wv   = t >> 5;
    const int tM   = (wv >> 1) << 4;     // 0 or 16 (cdd word l)
    const int tN   = (wv & 1) << 4;      // 0 or 16 (his word m)
    const int mrow = tM + (lane & 15);
    const int nrow = tN + (lane & 15);
    const int kOff = (lane >> 4) << 1;   // lanes 16-31 carry K+2,K+3

    // Loop-invariant padding: padded rows read a shared zero row.
    const float* __restrict__ aP = (mrow < L_) ? &sC[mrow][kOff] : &zrow[kOff];
    const float* __restrict__ bP = (nrow < L_) ? &sH[nrow][kOff] : &zrow[kOff];

    v8f acc = {};
    #pragma unroll 5
    for (int k = 0; k < D_; k += 4) {
        v2f a, bb;
        a.x  = aP[k];
        a.y  = aP[k + 1];
        bb.x = bP[k];
        bb.y = bP[k + 1];
        acc = __builtin_amdgcn_wmma_f32_16x16x4_f32(
                  /*neg_a=*/false, a, /*neg_b=*/false, bb,
                  /*c_mod=*/(short)0, acc, /*reuse_a=*/false, /*reuse_b=*/false);
    }

    // Spill C fragment (VGPR r: lanes 0-15 -> M=r, lanes 16-31 -> M=r+8).
    {
        const int col   = tN + (lane & 15);
        const int rbase = tM + ((lane >> 4) << 3);
        #pragma unroll
        for (int r = 0; r < 8; ++r) fus[rbase + r][col] = acc[r];
    }
    __syncthreads();

    // --- RBF kernel pooling: sum_m exp(-(x-mu)^2/(2s^2)) * mask_his,
    //     then log*mask_cdd*0.01 per (l,k) ---
    for (int idx = t; idx < L_ * K_; idx += 128) {
        const int   l      = idx / K_;
        const int   k      = idx % K_;
        const float mu     = -0.9f + 0.1f * (float)k;
        const float inv2s2 = (k == K_ - 1) ? (1.0f / (2.0f * 0.001f * 0.001f))
                                           : (1.0f / (2.0f * 0.1f * 0.1f));
        float s = 0.0f;
        for (int m = 0; m < L_; ++m) {
            float d = fus[l][m] - mu;
            s += expf(-(d * d) * inv2s2) * sHP[m];
        }
        psum[idx] = logf(fmaxf(s, 1e-10f)) * sCP[l] * 0.01f;
    }
    __syncthreads();

    // sum over l -> pooling_vectors[b,c,h,k]
    if (t < K_) {
        float s = 0.0f;
        for (int l = 0; l < L_; ++l) s += psum[l * K_ + t];
        pv[(size_t)blk * K_ + t] = s;
    }
}

// ---------------- Stage 3: attention + score + log_softmax -------------------
__global__ void knrm_attn_score(const float* __restrict__ pv,
                                const float* __restrict__ query,
                                const float* __restrict__ w,
                                const float* __restrict__ bias,
                                float* __restrict__ out)
{
    __shared__ float logits[HIS_];
    __shared__ float pooled[K_];
    __shared__ float sc[CDD_];

    const int b = blockIdx.x;
    const int t = threadIdx.x;
    const float scale = 1.0f / sqrtf((float)D_);

    for (int c = 0; c < CDD_; ++c) {
        const float* p = pv + (size_t)(b * CDD_ + c) * HIS_ * K_;
        if (t < HIS_) {
            float s = 0.0f;
            for (int k = 0; k < K_; ++k) s += query[k] * p[t * K_ + k];
            logits[t] = s * scale;
        }
        __syncthreads();
        if (t == 0) {   // softmax over HIS (50 values, trivial)
            float mx = logits[0];
            for (int hh = 1; hh < HIS_; ++hh) mx = fmaxf(mx, logits[hh]);
            float sum = 0.0f;
            for (int hh = 0; hh < HIS_; ++hh) { float e = expf(logits[hh] - mx); logits[hh] = e; sum += e; }
            float inv = 1.0f / sum;
            for (int hh = 0; hh < HIS_; ++hh) logits[hh] *= inv;
        }
        __syncthreads();
        if (t < K_) {   // pooled_k * ltr_w_k folded together
            float s = 0.0f;
            for (int hh = 0; hh < HIS_; ++hh) s += logits[hh] * p[hh * K_ + t];
            pooled[t] = s * w[t];
        }
        __syncthreads();
        if (t == 0) {
            float s = bias[0];
            for (int k = 0; k < K_; ++k) s += pooled[k];
            sc[c] = s;
        }
        __syncthreads();
    }
    if (t == 0) {       // log_softmax over CDD
        float mx = sc[0];
        for (int c = 1; c < CDD_; ++c) mx = fmaxf(mx, sc[c]);
        float sum = 0.0f;
        for (int c = 0; c < CDD_; ++c) sum += expf(sc[c] - mx);
        float lse = mx + logf(sum);
        for (int c = 0; c < CDD_; ++c) out[b * CDD_ + c] = sc[c] - lse;
    }
}

// ---------------- Host launch -----------------------------------------------
extern "C" void kernel_launch(void* const* d_in, const int* in_sizes, int n_in,
                              void* d_out, int out_size, void* d_ws, size_t ws_size,
                              hipStream_t stream)
{
    const int*   cand = (const int*)  d_in[0];   // (B,CDD,L)
    const int*   clk  = (const int*)  d_in[1];   // (B,HIS,L)
    const float* cpad = (const float*)d_in[2];   // (B,CDD,L)
    const float* hpad = (const float*)d_in[3];   // (B,HIS,L)
    const float* emb  = (const float*)d_in[4];   // (VOCAB,300)
    const float* qry  = (const float*)d_in[5];   // (1,K)
    const float* lw   = (const float*)d_in[6];   // (1,K)
    const float* lb   = (const float*)d_in[7];   // (1,)
    float*       out  = (float*)      d_out;     // (B,CDD) = 160

    // Workspace layout (fp32):
    //   cddN : 3200  x RS_
    //   hisN : 32000 x RS_
    //   pv   : 8000  x K_
    char* ws = (char*)d_ws;
    float* cddN = (float*)ws;
    float* hisN = cddN + (size_t)(B_ * CDD_ * L_) * RS_;
    float* pv   = hisN + (size_t)(B_ * HIS_ * L_) * RS_;

    // Stage 1: 35200 rows, one wave per row, 8 waves per 256-thread block.
    {
        int rows = B_ * CDD_ * L_ + B_ * HIS_ * L_;   // 35200
        int grid = (rows + 7) / 8;                    // 4400
        knrm_normalize<<<grid, 256, 0, stream>>>(cand, clk, emb, cddN, hisN);
    }
    // Stage 2: one block per (b,c,h) = 8000 blocks of 128 threads.
    knrm_fusion_pool<<<B_ * CDD_ * HIS_, 128, 0, stream>>>(cddN, hisN, cpad, hpad, pv);
    // Stage 3: one block per batch element.
    knrm_attn_score<<<B_, 64, 0, stream>>>(pv, qry, lw, lb, out);
}